// ModulatedConv2d_36541581754436
// MI455X (gfx1250) — compile-verified
//
#include <hip/hip_runtime.h>

#define BATCH 16
#define CIN   512
#define COUT  512
#define HH    64
#define WW    64
#define WDIM  512

#define BM 64            // c_out tile
#define CB 32            // c_in block per LDS stage
#define WPITCH 34        // padded ci pitch for weight LDS (even -> aligned pairs)
#define XPITCH 272       // 4 rows * 68 padded cols per ci
#define SW_ELEMS (9*BM*WPITCH)   // 19584
#define SX_ELEMS (CB*XPITCH)     // 8704

typedef float v2f __attribute__((ext_vector_type(2)));
typedef float v8f __attribute__((ext_vector_type(8)));

// ---------------- prologue kernels ----------------

__global__ void style_kernel(const float* __restrict__ w_embs,
                             const float* __restrict__ style_W,
                             const float* __restrict__ style_b,
                             float* __restrict__ style_out) {
    const int b = blockIdx.x;
    const float* we = w_embs + b * WDIM;
    for (int ci = threadIdx.x; ci < CIN; ci += blockDim.x) {
        const float* sw = style_W + ci * WDIM;
        float s = 0.f;
        for (int k = 0; k < WDIM; ++k) s += we[k] * sw[k];
        style_out[b * CIN + ci] = s + style_b[ci] + 1.0f;
    }
}

__global__ void wsq_kernel(const float* __restrict__ w, float* __restrict__ wsq) {
    const int idx = blockIdx.x * blockDim.x + threadIdx.x;   // co*512 + ci
    if (idx < COUT * CIN) {
        const float* p = w + idx * 9;
        float s = 0.f;
        #pragma unroll
        for (int t = 0; t < 9; ++t) s += p[t] * p[t];
        wsq[idx] = s;
    }
}

__global__ void demod_kernel(const float* __restrict__ style,
                             const float* __restrict__ wsq,
                             float* __restrict__ demod) {
    const int b  = blockIdx.x >> 9;
    const int co = blockIdx.x & 511;
    float s = 0.f;
    for (int ci = threadIdx.x; ci < CIN; ci += 64) {
        const float st = style[b * CIN + ci];
        s += st * st * wsq[co * CIN + ci];
    }
    __shared__ float red[64];
    red[threadIdx.x] = s;
    __syncthreads();
    for (int off = 32; off > 0; off >>= 1) {
        if (threadIdx.x < off) red[threadIdx.x] += red[threadIdx.x + off];
        __syncthreads();
    }
    if (threadIdx.x == 0) demod[blockIdx.x] = rsqrtf(red[0]);
}

// ---------------- main modulated-conv implicit GEMM ----------------
// grid: x = 32 row-pairs, y = 8 c_out blocks (64 each), z = 16 samples
// block: 256 threads = 8 wave32
// wave w: m-group = w&1 (m-tiles 2g..2g+1), n-group = w>>1 (n-tiles 2g..2g+1)

__global__ __launch_bounds__(256, 2)
void modconv_wmma(const float* __restrict__ imgs,
                  const float* __restrict__ weight,
                  const float* __restrict__ bias,
                  const float* __restrict__ style,
                  const float* __restrict__ demod,
                  float* __restrict__ out) {
    extern __shared__ float lds[];
    float* sW = lds;                          // [9][64][34]
    float* sX = lds + SW_ELEMS;               // [32][4][68]
    float* sS = lds + SW_ELEMS + SX_ELEMS;    // [512] styles of this sample

    const unsigned sWbyte = (unsigned)(size_t)(void*)sW;  // LDS byte offset (aperture in [63:32])

    const int tid    = threadIdx.x;
    const int lane   = tid & 31;
    const int wv     = tid >> 5;
    const int laneLo = lane & 15;
    const int hi     = lane >> 4;
    const int mg     = wv & 1;                // m-group
    const int ng     = wv >> 1;               // n-group 0..3

    const int b   = blockIdx.z;
    const int co0 = blockIdx.y * BM;
    const int r0  = blockIdx.x * 2;           // output rows r0, r0+1

    for (int i = tid; i < CIN; i += 256) sS[i] = style[b * CIN + i];
    __syncthreads();

    v8f acc[2][2] = {};

    for (int cib = 0; cib < CIN / CB; ++cib) {
        const int ci0 = cib * CB;
        __syncthreads();   // previous compute done before LDS overwrite

        // ---- Input halo tile: 32 ci x 4 rows x 68 cols, style folded in.
        // s = tid + i*256 over [ci][tr][j]; incremental decomposition (256 = 3*68+52).
        {
            int tr = tid / 68;
            int j  = tid - tr * 68;
            int ci = 0;
            int m  = tr * 64 + j;                // ci*4096 + tr*64 + j
            unsigned sidx = (unsigned)tid;
            const int roff = r0 * 64 - 65;
            const float* plane = imgs + (size_t)(b * CIN + ci0) * (HH * WW);
            #pragma unroll 2
            for (int i = 0; i < SX_ELEMS / 256; ++i) {
                const int gr = r0 - 1 + tr;
                const int gc = j - 1;
                float v = 0.f;
                if (((unsigned)gr < 64u) & ((unsigned)gc < 64u))
                    v = plane[m + roff] * sS[ci0 + ci];
                sX[sidx] = v;
                sidx += 256;
                j += 52; tr += 3; m += 244;
                if (j >= 68) { j -= 68; tr += 1; m -= 4; }
                if (tr >= 4) { tr -= 4; ci += 1; m += 3840; }
            }
        }

        // ---- Weight block: 9 taps x 64 co x 32 ci, async global->LDS scatter copy.
        // e = tid + i*256 over (co, ci, t); 256 = 28*9 + 4.
        {
            int wci = tid / 9;
            int wt  = tid - wci * 9;
            int gco = 0;
            unsigned ldsf = (unsigned)(wt * (BM * WPITCH) + wci);   // + gco*WPITCH
            int goff = tid;
            const int wbase0 = co0 * (CIN * 9) + ci0 * 9;
            #pragma unroll 4
            for (int i = 0; i < (9 * BM * CB) / 256; ++i) {
                const float* gp = weight + wbase0 + goff + gco * (CIN * 9 - CB * 9);
                const unsigned ldsaddr = sWbyte + ldsf * 4u;
                asm volatile("global_load_async_to_lds_b32 %0, %1, off"
                             :: "v"(ldsaddr), "v"(gp) : "memory");
                goff += 256;
                wt += 4; wci += 28; ldsf += 4u * (BM * WPITCH) + 28u;
                if (wt >= 9)   { wt -= 9;   wci += 1; ldsf += 1u - 9u * (BM * WPITCH); }
                if (wci >= CB) { wci -= CB; gco += 1; ldsf += (unsigned)(WPITCH - CB); }
            }
            asm volatile("s_wait_asynccnt 0x0" ::: "memory");
        }
        __syncthreads();

        // ---- 9 shifted GEMMs over this ci block
        for (int t = 0; t < 9; ++t) {
            const int ky = t / 3;
            const int kx = t - 3 * ky;
            const float* wb = sW + t * (BM * WPITCH) + laneLo * WPITCH + hi * 2;
            #pragma unroll
            for (int ks = 0; ks < 8; ++ks) {
                const int k0 = ks * 4;
                v2f a[2];
                #pragma unroll
                for (int mi = 0; mi < 2; ++mi) {
                    const float* ap = wb + (2 * mg + mi) * (16 * WPITCH) + k0;
                    a[mi].x = ap[0];
                    a[mi].y = ap[1];
                }
                v2f bf[2];
                #pragma unroll
                for (int ni = 0; ni < 2; ++ni) {
                    const int nt   = 2 * ng + ni;
                    const int orr  = nt >> 2;
                    const int colb = (nt & 3) * 16;
                    const int xi   = (k0 + hi * 2) * XPITCH + (orr + ky) * 68
                                     + colb + laneLo + kx;
                    bf[ni].x = sX[xi];
                    bf[ni].y = sX[xi + XPITCH];
                }
                #pragma unroll
                for (int mi = 0; mi < 2; ++mi)
                    #pragma unroll
                    for (int ni = 0; ni < 2; ++ni)
                        acc[mi][ni] = __builtin_amdgcn_wmma_f32_16x16x4_f32(
                            false, a[mi], false, bf[ni], (short)0, acc[mi][ni],
                            false, false);
            }
        }
    }

    // ---- Epilogue: demodulate + bias, write fp32 NCHW
    #pragma unroll
    for (int mi = 0; mi < 2; ++mi) {
        #pragma unroll
        for (int v = 0; v < 8; ++v) {
            const int co = co0 + (2 * mg + mi) * 16 + hi * 8 + v;
            const float dm = demod[b * COUT + co];
            const float bi = bias[co];
            #pragma unroll
            for (int ni = 0; ni < 2; ++ni) {
                const int nt  = 2 * ng + ni;
                const int row = r0 + (nt >> 2);
                const int col = (nt & 3) * 16 + laneLo;
                out[((b * COUT + co) * HH + row) * WW + col] = acc[mi][ni][v] * dm + bi;
            }
        }
    }
}

// ---------------- launcher ----------------

extern "C" void kernel_launch(void* const* d_in, const int* in_sizes, int n_in,
                              void* d_out, int out_size, void* d_ws, size_t ws_size,
                              hipStream_t stream) {
    const float* imgs        = (const float*)d_in[0];
    const float* w_embs      = (const float*)d_in[1];
    const float* conv_weight = (const float*)d_in[2];
    const float* conv_bias   = (const float*)d_in[3];
    const float* style_W     = (const float*)d_in[4];
    const float* style_b     = (const float*)d_in[5];
    float* outp = (float*)d_out;

    float* ws    = (float*)d_ws;
    float* style = ws;                   // 16*512
    float* demod = ws + BATCH * CIN;     // 16*512
    float* wsq   = ws + 2 * BATCH * CIN; // 512*512

    style_kernel<<<BATCH, 256, 0, stream>>>(w_embs, style_W, style_b, style);
    wsq_kernel<<<(COUT * CIN + 255) / 256, 256, 0, stream>>>(conv_weight, wsq);
    demod_kernel<<<BATCH * COUT, 64, 0, stream>>>(style, wsq, demod);

    dim3 grid(WW / 2, COUT / BM, BATCH);                 // 32 x 8 x 16
    size_t ldsB = (size_t)(SW_ELEMS + SX_ELEMS + CIN) * sizeof(float); // ~115 KB
    modconv_wmma<<<grid, 256, ldsB, stream>>>(imgs, conv_weight, conv_bias,
                                              style, demod, outp);
}